// EdgeConv_62302795596749
// MI455X (gfx1250) — compile-verified
//
#include <hip/hip_runtime.h>
#include <hip/hip_bf16.h>
#include <math.h>

typedef float v2f __attribute__((ext_vector_type(2)));
typedef float v8f __attribute__((ext_vector_type(8)));

#define TILES_PER_WAVE 8
#define EDGES_PER_BLOCK 512   // 4 waves * 8 tiles * 16 edges

// ---------------------------------------------------------------------------
// Kernel 1: h1 = [xi, xj-xi] @ W1 + b1   (E x 128) @ (128 x 64)
// also emits per-block per-channel sum / sumsq partials for BN1.
// ---------------------------------------------------------------------------
__global__ __launch_bounds__(128) void k1_gemm1(
    const float* __restrict__ x,
    const int*   __restrict__ erow,
    const int*   __restrict__ ecol,
    const float* __restrict__ W1,
    const float* __restrict__ b1,
    float*       __restrict__ h,
    float*       __restrict__ partial,
    long long E)
{
    // B fragments pre-swizzled: W1s[kk][nt][lane] = {B(2*kh,col), B(2*kh+1,col)}
    __shared__ __align__(16) float W1s[32][4][32][2];       // 32 KB
    __shared__ __align__(16) float feats[4][16][128];       // 32 KB (per-wave 16x128 A tiles)
    __shared__ __align__(16) float b1s[64];
    __shared__ float wsum[4][64];
    __shared__ float wsq[4][64];

    const int t    = threadIdx.x;
    const int wave = t >> 5;
    const int lane = t & 31;
    const int kh   = lane >> 4;   // half: 0 or 1
    const int ln   = lane & 15;

    // stage W1 fragments (row-major W1 is [128][64])
    for (int idx = t; idx < 32 * 4 * 32; idx += 128) {
        int l  = idx & 31;
        int nt = (idx >> 5) & 3;
        int kk = idx >> 7;
        int r0 = kk * 4 + 2 * (l >> 4);
        int c  = nt * 16 + (l & 15);
        W1s[kk][nt][l][0] = W1[r0 * 64 + c];
        W1s[kk][nt][l][1] = W1[(r0 + 1) * 64 + c];
    }
    if (t < 64) b1s[t] = b1[t];
    __syncthreads();

    float ls[4] = {0.f, 0.f, 0.f, 0.f};
    float lq[4] = {0.f, 0.f, 0.f, 0.f};

    const long long blockBase = (long long)blockIdx.x * EDGES_PER_BLOCK;

    for (int tile = 0; tile < TILES_PER_WAVE; ++tile) {
        const long long tileBase =
            blockBase + (long long)(wave * TILES_PER_WAVE + tile) * 16;

        // ---- stage A: feat = [xi, xj-xi], 16 edges x 128 ----
        {
            long long e = tileBase + ln;
            if (e >= E) e = E - 1;
            const int ri = erow[e];
            const int ci = ecol[e];
            const float4* xr = (const float4*)(x + (long long)ri * 64) + kh * 8;
            const float4* xc = (const float4*)(x + (long long)ci * 64) + kh * 8;
            float4* fi = (float4*)&feats[wave][ln][kh * 32];
            float4* fd = (float4*)&feats[wave][ln][64 + kh * 32];
#pragma unroll
            for (int q = 0; q < 8; ++q) {
                float4 a = xr[q];
                float4 b = xc[q];
                fi[q] = a;
                float4 d;
                d.x = b.x - a.x; d.y = b.y - a.y;
                d.z = b.z - a.z; d.w = b.w - a.w;
                fd[q] = d;
            }
        }
        // cross-lane visibility for this wave's freshly staged A tile.
        // (feats[wave] is wave-private; same-wave DS ops are in-order, so no
        //  trailing barrier is needed at the bottom of the tile loop.)
        __syncthreads();

        // ---- accumulators: init with bias (bias depends on channel only) ----
        v8f acc[4];
#pragma unroll
        for (int nt = 0; nt < 4; ++nt) {
            float bv = b1s[nt * 16 + ln];
#pragma unroll
            for (int v = 0; v < 8; ++v) acc[nt][v] = bv;
        }

        const float* frow = &feats[wave][ln][0];   // A row for M = ln
#pragma unroll 4
        for (int kk = 0; kk < 32; ++kk) {
            v2f a;
            a.x = frow[kk * 4 + 2 * kh];
            a.y = frow[kk * 4 + 2 * kh + 1];
#pragma unroll
            for (int nt = 0; nt < 4; ++nt) {
                v2f b;
                b.x = W1s[kk][nt][lane][0];
                b.y = W1s[kk][nt][lane][1];
                acc[nt] = __builtin_amdgcn_wmma_f32_16x16x4_f32(
                    false, a, false, b, (short)0, acc[nt], false, false);
            }
        }

        // ---- store h1 tile + accumulate BN1 stats ----
        // C/D layout: row M = v + 8*(lane>=16), col N = nt*16 + (lane&15)
        if (tileBase + 16 <= E) {         // wave-uniform fast path (no exec juggling)
            float* op0 = h + (tileBase + 8 * kh) * 64 + ln;
#pragma unroll
            for (int v = 0; v < 8; ++v) {
                float* op = op0 + (long long)v * 64;
#pragma unroll
                for (int nt = 0; nt < 4; ++nt) {
                    float val = acc[nt][v];
                    op[nt * 16] = val;
                    ls[nt] += val;
                    lq[nt] += val * val;
                }
            }
        } else {
#pragma unroll
            for (int v = 0; v < 8; ++v) {
                long long m = tileBase + v + 8 * kh;
                if (m < E) {
                    float* op = h + m * 64 + ln;
#pragma unroll
                    for (int nt = 0; nt < 4; ++nt) {
                        float val = acc[nt][v];
                        op[nt * 16] = val;
                        ls[nt] += val;
                        lq[nt] += val * val;
                    }
                }
            }
        }
    }

    // lanes L and L^16 hold the same channel -> fold halves, then fold waves
#pragma unroll
    for (int nt = 0; nt < 4; ++nt) {
        ls[nt] += __shfl_xor(ls[nt], 16, 32);
        lq[nt] += __shfl_xor(lq[nt], 16, 32);
    }
    if (kh == 0) {
#pragma unroll
        for (int nt = 0; nt < 4; ++nt) {
            wsum[wave][nt * 16 + ln] = ls[nt];
            wsq[wave][nt * 16 + ln]  = lq[nt];
        }
    }
    __syncthreads();
    if (t < 64) {
        float s = wsum[0][t] + wsum[1][t] + wsum[2][t] + wsum[3][t];
        float q = wsq[0][t] + wsq[1][t] + wsq[2][t] + wsq[3][t];
        partial[(long long)blockIdx.x * 128 + t]      = s;
        partial[(long long)blockIdx.x * 128 + 64 + t] = q;
    }
}

// ---------------------------------------------------------------------------
// Stats reduce: partials -> fused BN scale/bias   (single block, 128 threads)
// scaleBias[c] = g*rsqrt(var+eps), scaleBias[64+c] = be - mean*scale
// ---------------------------------------------------------------------------
__global__ __launch_bounds__(128) void k_stats(
    const float* __restrict__ partial, int nblocks, float invE,
    const float* __restrict__ g, const float* __restrict__ be,
    float* __restrict__ scaleBias)
{
    __shared__ float red[128];
    const int t = threadIdx.x;
    float s = 0.f;
    for (int b = 0; b < nblocks; ++b)
        s += partial[(long long)b * 128 + t];
    red[t] = s;
    __syncthreads();
    if (t < 64) {
        float mean = red[t] * invE;
        float var  = red[64 + t] * invE - mean * mean;
        float sc   = g[t] * rsqrtf(var + 1e-5f);
        scaleBias[t]      = sc;
        scaleBias[64 + t] = be[t] - mean * sc;
    }
}

// ---------------------------------------------------------------------------
// Kernel 3: h2 = relu(bn1(h1)) @ W2 + b2   in place on h  (E x 64)@(64 x 64)
// emits BN2 partials.
// ---------------------------------------------------------------------------
__global__ __launch_bounds__(128) void k3_gemm2(
    float*       __restrict__ h,        // in: h1_pre  out: h2_pre (in place)
    const float* __restrict__ W2,
    const float* __restrict__ b2,
    const float* __restrict__ sb1,      // [scale(64), bias(64)]
    float*       __restrict__ partial,
    long long E)
{
    __shared__ __align__(16) float W2s[16][4][32][2];       // 16 KB
    __shared__ __align__(16) float feats[4][16][64];        // 16 KB
    __shared__ __align__(16) float scs[64];
    __shared__ __align__(16) float bis[64];
    __shared__ __align__(16) float b2s[64];
    __shared__ float wsum[4][64];
    __shared__ float wsq[4][64];

    const int t    = threadIdx.x;
    const int wave = t >> 5;
    const int lane = t & 31;
    const int kh   = lane >> 4;
    const int ln   = lane & 15;

    for (int idx = t; idx < 16 * 4 * 32; idx += 128) {
        int l  = idx & 31;
        int nt = (idx >> 5) & 3;
        int kk = idx >> 7;
        int r0 = kk * 4 + 2 * (l >> 4);
        int c  = nt * 16 + (l & 15);
        W2s[kk][nt][l][0] = W2[r0 * 64 + c];
        W2s[kk][nt][l][1] = W2[(r0 + 1) * 64 + c];
    }
    if (t < 64) {
        scs[t] = sb1[t];
        bis[t] = sb1[64 + t];
        b2s[t] = b2[t];
    }
    __syncthreads();

    float ls[4] = {0.f, 0.f, 0.f, 0.f};
    float lq[4] = {0.f, 0.f, 0.f, 0.f};

    const long long blockBase = (long long)blockIdx.x * EDGES_PER_BLOCK;

    for (int tile = 0; tile < TILES_PER_WAVE; ++tile) {
        const long long tileBase =
            blockBase + (long long)(wave * TILES_PER_WAVE + tile) * 16;

        // ---- stage A = relu(scale*h1 + bias) ----
        {
            long long e = tileBase + ln;
            if (e >= E) e = E - 1;
            const float4* hr  = (const float4*)(h + e * 64) + kh * 8;
            const float4* sc4 = ((const float4*)scs) + kh * 8;
            const float4* bi4 = ((const float4*)bis) + kh * 8;
            float4* dst = (float4*)&feats[wave][ln][kh * 32];
#pragma unroll
            for (int q = 0; q < 8; ++q) {
                float4 v = hr[q];
                float4 s = sc4[q];
                float4 b = bi4[q];
                v.x = fmaxf(v.x * s.x + b.x, 0.f);
                v.y = fmaxf(v.y * s.y + b.y, 0.f);
                v.z = fmaxf(v.z * s.z + b.z, 0.f);
                v.w = fmaxf(v.w * s.w + b.w, 0.f);
                dst[q] = v;
            }
        }
        __syncthreads();

        v8f acc[4];
#pragma unroll
        for (int nt = 0; nt < 4; ++nt) {
            float bv = b2s[nt * 16 + ln];
#pragma unroll
            for (int v = 0; v < 8; ++v) acc[nt][v] = bv;
        }

        const float* frow = &feats[wave][ln][0];
#pragma unroll 4
        for (int kk = 0; kk < 16; ++kk) {
            v2f a;
            a.x = frow[kk * 4 + 2 * kh];
            a.y = frow[kk * 4 + 2 * kh + 1];
#pragma unroll
            for (int nt = 0; nt < 4; ++nt) {
                v2f b;
                b.x = W2s[kk][nt][lane][0];
                b.y = W2s[kk][nt][lane][1];
                acc[nt] = __builtin_amdgcn_wmma_f32_16x16x4_f32(
                    false, a, false, b, (short)0, acc[nt], false, false);
            }
        }

        if (tileBase + 16 <= E) {         // wave-uniform fast path
            float* op0 = h + (tileBase + 8 * kh) * 64 + ln;
#pragma unroll
            for (int v = 0; v < 8; ++v) {
                float* op = op0 + (long long)v * 64;
#pragma unroll
                for (int nt = 0; nt < 4; ++nt) {
                    float val = acc[nt][v];
                    op[nt * 16] = val;
                    ls[nt] += val;
                    lq[nt] += val * val;
                }
            }
        } else {
#pragma unroll
            for (int v = 0; v < 8; ++v) {
                long long m = tileBase + v + 8 * kh;
                if (m < E) {
                    float* op = h + m * 64 + ln;
#pragma unroll
                    for (int nt = 0; nt < 4; ++nt) {
                        float val = acc[nt][v];
                        op[nt * 16] = val;
                        ls[nt] += val;
                        lq[nt] += val * val;
                    }
                }
            }
        }
    }

#pragma unroll
    for (int nt = 0; nt < 4; ++nt) {
        ls[nt] += __shfl_xor(ls[nt], 16, 32);
        lq[nt] += __shfl_xor(lq[nt], 16, 32);
    }
    if (kh == 0) {
#pragma unroll
        for (int nt = 0; nt < 4; ++nt) {
            wsum[wave][nt * 16 + ln] = ls[nt];
            wsq[wave][nt * 16 + ln]  = lq[nt];
        }
    }
    __syncthreads();
    if (t < 64) {
        float s = wsum[0][t] + wsum[1][t] + wsum[2][t] + wsum[3][t];
        float q = wsq[0][t] + wsq[1][t] + wsq[2][t] + wsq[3][t];
        partial[(long long)blockIdx.x * 128 + t]      = s;
        partial[(long long)blockIdx.x * 128 + 64 + t] = q;
    }
}

// ---------------------------------------------------------------------------
// Kernel 5: out[i,c] = max_k relu(bn2(h2[i*16+k, c]))
// (edge_row = repeat(arange(N),16) -> each node's edges are contiguous rows)
// ---------------------------------------------------------------------------
__global__ __launch_bounds__(256) void k5_out(
    const float* __restrict__ h,
    const float* __restrict__ sb2,
    float*       __restrict__ out,
    long long total)                     // N * 64
{
    long long idx = (long long)blockIdx.x * 256 + threadIdx.x;
    if (idx >= total) return;
    const long long node = idx >> 6;
    const int c = (int)(idx & 63);
    const float sc = sb2[c];
    const float bi = sb2[64 + c];
    const float* hp = h + node * 16 * 64 + c;
    float m = 0.0f;                      // relu output >= 0, so 0 is the identity
#pragma unroll
    for (int k = 0; k < 16; ++k) {
        float v = fmaxf(fmaf(hp[k * 64], sc, bi), 0.0f);
        m = fmaxf(m, v);
    }
    out[idx] = m;
}

// ---------------------------------------------------------------------------
extern "C" void kernel_launch(void* const* d_in, const int* in_sizes, int n_in,
                              void* d_out, int out_size, void* d_ws, size_t ws_size,
                              hipStream_t stream)
{
    const float* x   = (const float*)d_in[0];
    const int*   erow = (const int*)d_in[1];
    const int*   ecol = (const int*)d_in[2];
    const float* W1  = (const float*)d_in[3];
    const float* b1  = (const float*)d_in[4];
    const float* g1  = (const float*)d_in[5];
    const float* be1 = (const float*)d_in[6];
    const float* W2  = (const float*)d_in[7];
    const float* b2  = (const float*)d_in[8];
    const float* g2  = (const float*)d_in[9];
    const float* be2 = (const float*)d_in[10];
    float* out = (float*)d_out;

    const long long Nn = (long long)in_sizes[0] / 64;
    const long long E  = (long long)in_sizes[1];
    const int nblocks  = (int)((E + EDGES_PER_BLOCK - 1) / EDGES_PER_BLOCK);

    // workspace layout (floats): [sb1(128)][sb2(128)][p1(nb*128)][p2(nb*128)][h(E*64)]
    float* ws  = (float*)d_ws;
    float* sb1 = ws;
    float* sb2 = ws + 128;
    float* p1  = ws + 256;
    float* p2  = p1 + (long long)nblocks * 128;
    long long hOff = 256 + 2LL * nblocks * 128;
    hOff = (hOff + 63) & ~63LL;          // keep h 256B-aligned for float4
    float* h = ws + hOff;

    const float invE = (float)(1.0 / (double)E);

    k1_gemm1<<<nblocks, 128, 0, stream>>>(x, erow, ecol, W1, b1, h, p1, E);
    k_stats<<<1, 128, 0, stream>>>(p1, nblocks, invE, g1, be1, sb1);
    k3_gemm2<<<nblocks, 128, 0, stream>>>(h, W2, b2, sb1, p2, E);
    k_stats<<<1, 128, 0, stream>>>(p2, nblocks, invE, g2, be2, sb2);

    const long long total = Nn * 64;
    k5_out<<<(int)((total + 255) / 256), 256, 0, stream>>>(h, sb2, out, total);
}